// FPSSampling_40810779246882
// MI455X (gfx1250) — compile-verified
//
#include <hip/hip_runtime.h>
#include <hip/hip_bf16.h>

#define FPS_B   32
#define FPS_N   100000
#define FPS_NG  2048
#define TPB     1024
#define NWAVES  (TPB / 32)          // 32
#define PPT     98                  // ceil(100000 / 1024)
#define INIT_DIST 1.0e10f

// Faithful fp32 squared distance: (x-c)^2 summed left-to-right, no FMA contraction.
__device__ __forceinline__ float sq_dist(float x, float y, float z,
                                         float cx, float cy, float cz) {
    float dx = __fsub_rn(x, cx);
    float dy = __fsub_rn(y, cy);
    float dz = __fsub_rn(z, cz);
    float a = __fmul_rn(dx, dx);
    float b = __fmul_rn(dy, dy);
    float c = __fmul_rn(dz, dz);
    return __fadd_rn(__fadd_rn(a, b), c);
}

__device__ __forceinline__ void argmax_combine(float& v, int& i, float ov, int oi) {
    // jnp.argmax semantics: max value, first (lowest) index on ties
    if (ov > v || (ov == v && oi < i)) { v = ov; i = oi; }
}

extern "C" __global__ void __launch_bounds__(TPB)
fps_kernel(const float* __restrict__ xyz,   // [B, N, 3]
           const int*   __restrict__ far0,  // [B]
           float*       __restrict__ out)   // [B, NUM_GROUP, 3]
{
    const int b   = blockIdx.x;
    const int tid = threadIdx.x;
    __builtin_assume(tid < TPB);

    const float* pts  = xyz + (size_t)b * FPS_N * 3;
    float*       outb = out + (size_t)b * FPS_NG * 3;

    __shared__ float s_c[3];
    __shared__ float s_val[NWAVES];
    __shared__ int   s_idx[NWAVES];

    // Running min-distance lives in VGPRs (fully unrolled constant indexing).
    // 98 floats/lane + scalar temps ~= 115 VGPRs: fits 128-reg cap at 8 waves/SIMD.
    float dist[PPT];
#pragma unroll
    for (int k = 0; k < PPT; ++k) dist[k] = INIT_DIST;

    // Warm L2 with this batch's points (gfx1250 global_prefetch_b8 path).
    // Clamped tail -> branchless immediate-offset prefetch run.
#pragma unroll
    for (int k = 0; k < PPT; ++k) {
        int idx = tid + k * TPB;
        if (k == PPT - 1) idx = min(idx, FPS_N - 1);
        __builtin_prefetch(pts + 3 * idx, 0, 1);
    }

    // Seed: record initial centroid (pre-update, matches scan output order).
    if (tid == 0) {
        int f = far0[b];
        float cx = pts[3 * f + 0];
        float cy = pts[3 * f + 1];
        float cz = pts[3 * f + 2];
        s_c[0] = cx; s_c[1] = cy; s_c[2] = cz;
        outb[0] = cx; outb[1] = cy; outb[2] = cz;
    }
    __syncthreads();

    // Serial FPS scan: exactly one static copy of the body.
#pragma unroll 1
    for (int t = 0; t < FPS_NG; ++t) {
        const float cx = s_c[0], cy = s_c[1], cz = s_c[2];

        // Sweep (proven no-spill structure).
        float best = -1.0f;
        int   bidx = 0x7fffffff;
#pragma unroll
        for (int k = 0; k < PPT; ++k) {
            int idx = tid + k * TPB;
            if (idx < FPS_N) {
                const float* p = pts + 3 * idx;
                float d  = sq_dist(p[0], p[1], p[2], cx, cy, cz);
                float nd = fminf(dist[k], d);
                dist[k] = nd;
                // strict > keeps the earliest index within this thread's
                // ascending strided sequence
                if (nd > best) { best = nd; bidx = idx; }
            }
        }

        // Wave32 argmax reduction.
#pragma unroll
        for (int off = 16; off > 0; off >>= 1) {
            float ov = __shfl_down(best, (unsigned)off, 32);
            int   oi = __shfl_down(bidx, (unsigned)off, 32);
            argmax_combine(best, bidx, ov, oi);
        }

        const int wave = tid >> 5;
        const int lane = tid & 31;
        if (lane == 0) { s_val[wave] = best; s_idx[wave] = bidx; }
        __syncthreads();

        // Wave 0 finishes the reduction; its lane 0 immediately publishes the
        // new centroid (LDS broadcast + output record for step t+1).
        if (wave == 0) {
            best = s_val[lane];
            bidx = s_idx[lane];
#pragma unroll
            for (int off = 16; off > 0; off >>= 1) {
                float ov = __shfl_down(best, (unsigned)off, 32);
                int   oi = __shfl_down(bidx, (unsigned)off, 32);
                argmax_combine(best, bidx, ov, oi);
            }
            if (lane == 0) {
                const int f = bidx;
                float ncx = pts[3 * f + 0];
                float ncy = pts[3 * f + 1];
                float ncz = pts[3 * f + 2];
                s_c[0] = ncx; s_c[1] = ncy; s_c[2] = ncz;
                if (t + 1 < FPS_NG) {
                    outb[3 * (t + 1) + 0] = ncx;
                    outb[3 * (t + 1) + 1] = ncy;
                    outb[3 * (t + 1) + 2] = ncz;
                }
            }
        }
        __syncthreads();
    }
}

extern "C" void kernel_launch(void* const* d_in, const int* in_sizes, int n_in,
                              void* d_out, int out_size, void* d_ws, size_t ws_size,
                              hipStream_t stream) {
    (void)in_sizes; (void)n_in; (void)out_size; (void)d_ws; (void)ws_size;
    const float* xyz  = (const float*)d_in[0];   // [32, 100000, 3] fp32
    const int*   far0 = (const int*)d_in[1];     // [32] int
    float*       out  = (float*)d_out;           // [32, 2048, 3] fp32

    fps_kernel<<<FPS_B, TPB, 0, stream>>>(xyz, far0, out);
}